// ImportancePoolingLayer_28424093564961
// MI455X (gfx1250) — compile-verified
//
#include <hip/hip_runtime.h>

// CDNA5 / gfx1250, wave32. One wave per node.
//
// out[n,:] = sum_k wn[n,k] * x[neighbors[n,k],:],  wn = weights / sum (uniform
// fallback if sum==0).  N=50000, K=32, D=128.
//
// Memory-bound (x = 25.6MB fits in 192MB L2; the 820MB gather is L2-served).
// Matrix path: V_WMMA_F32_16X16X4_F32 with a row-replicated A operand:
//   A[m,j] = wn[k0+j]            (all 16 rows identical, exact f32)
//   B[j,c] = x[nbr[k0+j]][dim c]
//   D[m,c] = out[dim c] partial, accumulated over 8 k-chunks through C.
// Dim chunks are processed in pairs (two accumulators) with double-buffered
// B loads so ~32 loads are in flight while the 16-deep WMMA chain executes,
// and each pair finishes with one full-wave contiguous 128B store.

typedef float v2f __attribute__((ext_vector_type(2)));
typedef float v8f __attribute__((ext_vector_type(8)));

__global__ __launch_bounds__(256) void importance_pool_wmma_kernel(
    const float* __restrict__ x,          // [N,128]
    const float* __restrict__ weights,    // [N,32]
    const long long* __restrict__ nbrs,   // [N,32] int64
    float* __restrict__ out,              // [N,128]
    int N)
{
  const int lane = (int)(threadIdx.x & 31u);
  const int node = (int)blockIdx.x * 8 + (int)(threadIdx.x >> 5);
  if (node >= N) return;   // wave-uniform -> EXEC stays all-1s for WMMA

  // Lane k owns weight k and neighbor k (wave-uniform base -> coalesced).
  const float wk       = weights[(long long)node * 32 + lane];
  const long long idx  = nbrs[(long long)node * 32 + lane];
  const unsigned rowOff = (unsigned)idx * 512u;      // row byte offset (128*4B)

  // Wave-wide weight sum with zero-sum uniform fallback.
  float s = wk;
  #pragma unroll
  for (int d = 16; d >= 1; d >>= 1) s += __shfl_xor(s, d, 32);
  const float wn = (s == 0.0f) ? (1.0f / 32.0f) : (wk / s);

  // K-split of A(16x4)/B(4x16) across lane halves:
  // lanes 0-15 carry K={0,1}, lanes 16-31 carry K={2,3} (VGPR0/VGPR1).
  const int half2 = (lane >> 4) << 1;
  const int dl    = lane & 15;

  // Per k-chunk: replicated A weights + per-lane pointers into the two
  // gathered rows this lane reads; dim chunks become immediate offsets.
  v2f a[8];
  const float* p0[8];
  const float* p1[8];
  #pragma unroll
  for (int kc = 0; kc < 8; ++kc) {
    const int k0 = kc * 4 + half2;
    a[kc].x = __shfl(wn, k0,     32);
    a[kc].y = __shfl(wn, k0 + 1, 32);
    const unsigned o0 = (unsigned)__shfl((int)rowOff, k0,     32);
    const unsigned o1 = (unsigned)__shfl((int)rowOff, k0 + 1, 32);
    p0[kc] = (const float*)((const char*)x + o0) + dl;
    p1[kc] = (const float*)((const char*)x + o1) + dl;
  }

  // Double-buffered B operands for a pair of 16-dim chunks:
  // b[buf][kc]   = even chunk (dims pair*32 +  0..15)
  // b[buf][8+kc] = odd  chunk (dims pair*32 + 16..31)
  v2f b[2][16];

  #pragma unroll
  for (int kc = 0; kc < 8; ++kc) {          // preload pair 0 (dims 0..31)
    b[0][kc].x     = p0[kc][0];
    b[0][kc].y     = p1[kc][0];
    b[0][8 + kc].x = p0[kc][16];
    b[0][8 + kc].y = p1[kc][16];
  }

  #pragma unroll
  for (int p = 0; p < 4; ++p) {             // 4 pairs x 32 dims = 128
    const int cur = p & 1;
    if (p < 3) {                            // prefetch next pair's 32 loads
      const int nxt  = cur ^ 1;
      const int dofs = (p + 1) * 32;        // float offset (immediate)
      #pragma unroll
      for (int kc = 0; kc < 8; ++kc) {
        b[nxt][kc].x     = p0[kc][dofs];
        b[nxt][kc].y     = p1[kc][dofs];
        b[nxt][8 + kc].x = p0[kc][dofs + 16];
        b[nxt][8 + kc].y = p1[kc][dofs + 16];
      }
    }

    // Two independent accumulator chains, interleaved issue.
    v8f ce = {};
    v8f co = {};
    #pragma unroll
    for (int kc = 0; kc < 8; ++kc) {
      ce = __builtin_amdgcn_wmma_f32_16x16x4_f32(
              false, a[kc], false, b[cur][kc],     (short)0, ce, false, false);
      co = __builtin_amdgcn_wmma_f32_16x16x4_f32(
              false, a[kc], false, b[cur][8 + kc], (short)0, co, false, false);
    }

    // Every lane's VGPR0 of D replicates the chunk: ce[0] = out[p*32+(lane&15)],
    // co[0] = out[p*32+16+(lane&15)].  Lane halves write even/odd chunk ->
    // one full-wave contiguous 128B store, no EXEC juggling.
    const float v = (lane & 16) ? co[0] : ce[0];
    out[(long long)node * 128 + p * 32 + lane] = v;
  }
}

extern "C" void kernel_launch(void* const* d_in, const int* in_sizes, int n_in,
                              void* d_out, int out_size, void* d_ws, size_t ws_size,
                              hipStream_t stream) {
  const float*     x   = (const float*)d_in[0];
  const float*     w   = (const float*)d_in[1];
  const long long* nbr = (const long long*)d_in[2];
  float*           out = (float*)d_out;

  const int N = in_sizes[1] / 32;           // weights is [N,32]
  const int blocks = (N + 7) / 8;           // 8 wave32 per 256-thread block
  importance_pool_wmma_kernel<<<blocks, 256, 0, stream>>>(x, w, nbr, out, N);
}